// ConstraintFuser_6408091206348
// MI455X (gfx1250) — compile-verified
//
#include <hip/hip_runtime.h>

#define B_  2048
#define C_  50
#define D_  512
#define H_  51      // hidden dim (padded to 64 for WMMA)

typedef __attribute__((ext_vector_type(2))) float v2f;
typedef __attribute__((ext_vector_type(8))) float v8f;
typedef __attribute__((ext_vector_type(4))) int   v4i;

#define AS_GLOBAL __attribute__((address_space(1)))
#define AS_LOCAL  __attribute__((address_space(3)))

#if __has_builtin(__builtin_amdgcn_global_load_async_to_lds_b128)
#define HAVE_ASYNC_LDS 1
#else
#define HAVE_ASYNC_LDS 0
#endif

__device__ __forceinline__ int imin(int a, int b) { return a < b ? a : b; }

// ---------------------------------------------------------------------------
// Kernel 1: embedding gathers + head score + weighted pooling.
// One block per batch row b; 256 threads (8 waves).
// Phase A: one wave per constraint computes q . h_emb[c] (coalesced float4
//          row loads, single wave32 shfl reduction). Only 2 barriers total.
// Phase B: barrier-free pooling; prefetch next constraint's t/r rows
//          (global_prefetch_b8) to keep the random-gather stream deep.
// Writes pooled[b,:] into d_out (consumed in-place by kernel 2).
// ---------------------------------------------------------------------------
__global__ __launch_bounds__(256)
void cf_pool_kernel(const float* __restrict__ query,
                    const int*   __restrict__ heads,
                    const int*   __restrict__ tails,
                    const int*   __restrict__ rels,
                    const float* __restrict__ eemb,
                    const float* __restrict__ remb,
                    float*       __restrict__ pooled)
{
    const int b    = blockIdx.x;
    const int t    = threadIdx.x;
    const int lane = t & 31;
    const int wave = t >> 5;

    __shared__ __align__(16) float q_lds[D_];
    __shared__ float sc_lds[C_];

    // Stage query row into LDS (coalesced float2)
    *(float2*)(q_lds + 2 * t) = *(const float2*)(query + (size_t)b * D_ + 2 * t);
    __syncthreads();

    // ---- Phase A: head scores, one wave per constraint, round-robin ----
    for (int c = wave; c < C_; c += 8) {
        const int ih = heads[b * C_ + c];
        const float* hrow = eemb + (size_t)ih * D_;
        float p = 0.f;
        #pragma unroll
        for (int j = 0; j < 4; ++j) {
            const float4 hv = *(const float4*)(hrow  + 4 * lane + 128 * j);
            const float4 qv = *(const float4*)(q_lds + 4 * lane + 128 * j);
            p += hv.x * qv.x + hv.y * qv.y + hv.z * qv.z + hv.w * qv.w;
        }
        #pragma unroll
        for (int off = 16; off > 0; off >>= 1)
            p += __shfl_xor(p, off, 32);
        if (lane == 0) sc_lds[c] = p;
    }
    __syncthreads();

    // ---- Phase B: pooled accumulate, barrier-free, prefetch one ahead ----
    float accx = 0.f, accy = 0.f;
    int it_ = tails[b * C_];
    int ir  = rels [b * C_];
    for (int c = 0; c < C_; ++c) {
        const bool more = (c + 1 < C_);
        const int itn = more ? tails[b * C_ + c + 1] : it_;
        const int irn = more ? rels [b * C_ + c + 1] : ir;
        __builtin_prefetch(eemb + (size_t)itn * D_ + 2 * t, 0, 1);
        __builtin_prefetch(remb + (size_t)irn * D_ + 2 * t, 0, 1);

        const float s  = sc_lds[c];
        const float2 t2 = *(const float2*)(eemb + (size_t)it_ * D_ + 2 * t);
        const float2 r2 = *(const float2*)(remb + (size_t)ir  * D_ + 2 * t);
        accx += s * (t2.x + r2.x);
        accy += s * (t2.y + r2.y);
        it_ = itn; ir = irn;
    }
    float2 res; res.x = accx; res.y = accy;
    *(float2*)(pooled + (size_t)b * D_ + 2 * t) = res;
}

// ---------------------------------------------------------------------------
// Kernel 2: fused FFN  out = relu(pooled@w1 + b1) @ w2 + b2 + query
// via V_WMMA_F32_16X16X4_F32. 128 threads = 4 waves; one 16-row M-tile/block.
// A-tile (16x512) staged into LDS with async global->LDS (ASYNCcnt) so the
// A-fragment reads are LDS b64 loads instead of 2KB-strided global loads.
// All padded-region handling is branchless via address clamping: padded
// mid columns are zeroed after GEMM1, and padded K rows in GEMM2 multiply
// exact zeros, so garbage (finite) B values are harmless. EXEC stays all-1s.
// ---------------------------------------------------------------------------
__global__ __launch_bounds__(128)
void cf_ffn_kernel(float* data,                    // d_out: pooled in, result out
                   const float* __restrict__ query,
                   const float* __restrict__ w1, const float* __restrict__ b1,
                   const float* __restrict__ w2, const float* __restrict__ b2)
{
    const int tid  = threadIdx.x;
    const int wave = tid >> 5;        // 0..3
    const int lane = tid & 31;
    const int row  = lane & 15;       // M-row for A-frag, N-col for B/C/D frags
    const int hi   = lane >> 4;       // lane group 0/1
    const int koff = hi * 2;          // K-pair offset per ISA A/B layout
    const int mb   = blockIdx.x * 16;

    __shared__ __align__(16) float Atile[16 * D_];   // 32 KB pooled tile
    __shared__ float mid[16 * 64];                   // 4 KB hidden tile

    // ---- Stage A tile: 8192 floats = 2048 x 16B chunks; 16 per thread ----
    {
        const float* gsrc = data + (size_t)mb * D_;
        #pragma unroll
        for (int i = 0; i < 16; ++i) {
            const int idx = (i * 128 + tid) * 4;     // float index, 16B aligned
#if HAVE_ASYNC_LDS
            __builtin_amdgcn_global_load_async_to_lds_b128(
                (AS_GLOBAL v4i*)(gsrc + idx),
                (AS_LOCAL  v4i*)(Atile + idx), 0, 0);
#else
            *(float4*)(Atile + idx) = *(const float4*)(gsrc + idx);
#endif
        }
#if HAVE_ASYNC_LDS
        asm volatile("s_wait_asynccnt 0x0" ::: "memory");
#endif
    }
    __syncthreads();

    // ---------------- GEMM1: mid = relu(Atile @ w1 + b1) ----------------
    {
        const int nb1  = wave * 16;
        const int col  = nb1 + row;
        const int colc = imin(col, H_ - 1);          // clamped, always in-bounds
        const bool cv  = (col < H_);

        v8f acc = {};
        for (int k = 0; k < D_; k += 4) {
            v2f a;
            a.x = Atile[row * D_ + k + koff];
            a.y = Atile[row * D_ + k + koff + 1];
            v2f bf;
            bf.x = w1[(k + koff    ) * H_ + colc];   // branchless (clamped)
            bf.y = w1[(k + koff + 1) * H_ + colc];
            acc = __builtin_amdgcn_wmma_f32_16x16x4_f32(
                      false, a, false, bf, (short)0, acc, false, false);
        }
        const float bias = b1[colc];
        #pragma unroll
        for (int i = 0; i < 8; ++i) {
            const int m = i + 8 * hi;
            float v = acc[i] + bias;
            v = v > 0.f ? v : 0.f;
            v = cv ? v : 0.f;                        // zero padded hidden cols
            mid[m * 64 + col] = v;
        }
    }
    __syncthreads();

    // ---------------- GEMM2: out = mid @ w2 + b2 + query -----------------
    // Padded K rows (kk >= 51) read clamped w2 addresses; they multiply the
    // zeroed mid columns, contributing exactly 0.
    for (int it = 0; it < 8; ++it) {
        const int nb2 = (wave * 8 + it) * 16;
        const int n   = nb2 + row;
        v8f acc = {};
        #pragma unroll
        for (int k = 0; k < 64; k += 4) {
            const int kk = k + koff;
            v2f a;
            a.x = mid[row * 64 + kk];
            a.y = mid[row * 64 + kk + 1];
            v2f bf;
            bf.x = w2[(size_t)imin(kk,     H_ - 1) * D_ + n];
            bf.y = w2[(size_t)imin(kk + 1, H_ - 1) * D_ + n];
            acc = __builtin_amdgcn_wmma_f32_16x16x4_f32(
                      false, a, false, bf, (short)0, acc, false, false);
        }
        const float bias2 = b2[n];
        #pragma unroll
        for (int i = 0; i < 8; ++i) {
            const int m = mb + i + 8 * hi;
            data[(size_t)m * D_ + n] = acc[i] + bias2 + query[(size_t)m * D_ + n];
        }
    }
}

// ---------------------------------------------------------------------------
extern "C" void kernel_launch(void* const* d_in, const int* in_sizes, int n_in,
                              void* d_out, int out_size, void* d_ws, size_t ws_size,
                              hipStream_t stream)
{
    (void)in_sizes; (void)n_in; (void)out_size; (void)d_ws; (void)ws_size;

    const float* query = (const float*)d_in[0];
    const int*   heads = (const int*)  d_in[1];
    const int*   tails = (const int*)  d_in[2];
    const int*   rels  = (const int*)  d_in[3];
    const float* eemb  = (const float*)d_in[4];
    const float* remb  = (const float*)d_in[5];
    const float* w1    = (const float*)d_in[6];
    const float* b1    = (const float*)d_in[7];
    const float* w2    = (const float*)d_in[8];
    const float* b2    = (const float*)d_in[9];
    float* out = (float*)d_out;

    // Stage 1: pooled -> d_out (fully overwritten every call; deterministic)
    cf_pool_kernel<<<B_, 256, 0, stream>>>(query, heads, tails, rels, eemb, remb, out);
    // Stage 2: FFN in-place on d_out (per-block read-then-write of own rows)
    cf_ffn_kernel<<<B_ / 16, 128, 0, stream>>>(out, query, w1, b1, w2, b2);
}